// TopkHintMSELoss_463856468414
// MI455X (gfx1250) — compile-verified
//
#include <hip/hip_runtime.h>

typedef __attribute__((ext_vector_type(2))) float v2f;
typedef __attribute__((ext_vector_type(4))) float v4f;
typedef __attribute__((ext_vector_type(8))) float v8f;

constexpr int kB  = 32;
constexpr int kC  = 512;
constexpr int kHW = 64 * 64;   // 4096 spatial elements per (b,c)
constexpr int kK  = 256;       // TOPK * C

// ---------------------------------------------------------------------------
// Kernel 1: per (b,c) row, compute sum(targets^2) and sum((inputs-targets)^2).
// Streaming, HBM-bound: 512 MB total traffic -> ~22 us floor at 23.3 TB/s.
// Reduction tail uses V_WMMA_F32_16X16X4_F32 with an all-ones B matrix:
// summing all 16 output rows equals the sum of all 64 A-operand values,
// which is layout-permutation invariant, so the result is exact fp32.
// ---------------------------------------------------------------------------
__global__ __launch_bounds__(256)
void rowstats_kernel(const float* __restrict__ inputs,
                     const float* __restrict__ targets,
                     float* __restrict__ norm2,
                     float* __restrict__ ssd) {
    const int row  = blockIdx.x;                 // row = b*kC + c
    const int t    = threadIdx.x;                // 0..255, 8 waves
    const size_t base = (size_t)row * kHW;

    const v4f* ip = (const v4f*)(inputs  + base);
    const v4f* tp = (const v4f*)(targets + base);

    float st = 0.0f;   // sum targets^2
    float sd = 0.0f;   // sum (inputs - targets)^2
#pragma unroll
    for (int j = 0; j < 4; ++j) {
        v4f iv = __builtin_nontemporal_load(&ip[j * 256 + t]);
        v4f tv = __builtin_nontemporal_load(&tp[j * 256 + t]);
#pragma unroll
        for (int e = 0; e < 4; ++e) {
            float ti = tv[e];
            float di = iv[e] - ti;
            st = fmaf(ti, ti, st);
            sd = fmaf(di, di, sd);
        }
    }

    __shared__ float l_st[256];
    __shared__ float l_sd[256];
    l_st[t] = st;
    l_sd[t] = sd;
    __syncthreads();

    // Wave 0 only (wave-uniform branch -> EXEC all-1s inside, as WMMA requires)
    if (t < 32) {
        v2f ones; ones.x = 1.0f; ones.y = 1.0f;   // all-ones B (layout-invariant)
        v8f acc_t = {};
        v8f acc_d = {};
#pragma unroll
        for (int c2 = 0; c2 < 4; ++c2) {
            v2f a; a.x = l_st[c2 * 64 + t]; a.y = l_st[c2 * 64 + 32 + t];
            acc_t = __builtin_amdgcn_wmma_f32_16x16x4_f32(
                false, a, false, ones, (short)0, acc_t, false, false);
            v2f b; b.x = l_sd[c2 * 64 + t]; b.y = l_sd[c2 * 64 + 32 + t];
            acc_d = __builtin_amdgcn_wmma_f32_16x16x4_f32(
                false, b, false, ones, (short)0, acc_d, false, false);
        }
        // Sum the 8 D VGPRs (rows 0-7 on lanes 0-15, rows 8-15 on lanes 16-31),
        // then add the other lane-half -> total over all 256 partials.
        float rt = acc_t[0] + acc_t[1] + acc_t[2] + acc_t[3] +
                   acc_t[4] + acc_t[5] + acc_t[6] + acc_t[7];
        float rd = acc_d[0] + acc_d[1] + acc_d[2] + acc_d[3] +
                   acc_d[4] + acc_d[5] + acc_d[6] + acc_d[7];
        rt += __shfl_xor(rt, 16);
        rd += __shfl_xor(rd, 16);
        if (t == 0) {
            norm2[row] = rt;   // compare on norm^2: sqrt is monotonic, skip it
            ssd[row]   = rd;
        }
    }
}

// ---------------------------------------------------------------------------
// Kernel 2: per batch, select top-k channels by norm^2 (rank-by-counting with
// lowest-index tie-break, matching jax.lax.top_k), sum kept ssd values.
// ---------------------------------------------------------------------------
__global__ __launch_bounds__(512)
void topk_mask_kernel(const float* __restrict__ norm2,
                      const float* __restrict__ ssd,
                      float* __restrict__ partial) {
    const int b = blockIdx.x;
    const int c = threadIdx.x;   // 0..511

    __shared__ float n2[kC];
    __shared__ float red[kC];

    const float v = norm2[b * kC + c];
    n2[c] = v;
    __syncthreads();

    int rank = 0;
    for (int j = 0; j < kC; ++j) {
        float u = n2[j];
        rank += (u > v) || (u == v && j < c);
    }
    red[c] = (rank < kK) ? ssd[b * kC + c] : 0.0f;
    __syncthreads();

#pragma unroll
    for (int s = kC / 2; s > 0; s >>= 1) {
        if (c < s) red[c] += red[c + s];
        __syncthreads();
    }
    if (c == 0) partial[b] = red[0];
}

// ---------------------------------------------------------------------------
// Kernel 3: sum 32 per-batch partials, apply exact scale 1/(4096*8192) = 2^-25.
// ---------------------------------------------------------------------------
__global__ __launch_bounds__(32)
void final_kernel(const float* __restrict__ partial, float* __restrict__ out) {
    float v = partial[threadIdx.x];
#pragma unroll
    for (int off = 16; off > 0; off >>= 1) v += __shfl_xor(v, off);
    if (threadIdx.x == 0) out[0] = v * (1.0f / 33554432.0f);  // 2^-25 exact
}

// ---------------------------------------------------------------------------
extern "C" void kernel_launch(void* const* d_in, const int* in_sizes, int n_in,
                              void* d_out, int out_size, void* d_ws, size_t ws_size,
                              hipStream_t stream) {
    const float* inputs  = (const float*)d_in[0];
    const float* targets = (const float*)d_in[1];
    float* out = (float*)d_out;

    float* norm2   = (float*)d_ws;          // kB*kC floats
    float* ssd     = norm2 + kB * kC;       // kB*kC floats
    float* partial = ssd   + kB * kC;       // kB floats   (~131 KB total)

    rowstats_kernel<<<kB * kC, 256, 0, stream>>>(inputs, targets, norm2, ssd);
    topk_mask_kernel<<<kB, 512, 0, stream>>>(norm2, ssd, partial);
    final_kernel<<<1, 32, 0, stream>>>(partial, out);
}